// ChamferLoss_10368051052748
// MI455X (gfx1250) — compile-verified
//
#include <hip/hip_runtime.h>
#include <math.h>

typedef float v2f __attribute__((ext_vector_type(2)));
typedef float v8f __attribute__((ext_vector_type(8)));
typedef int   v4i __attribute__((ext_vector_type(4)));

#define AS1 __attribute__((address_space(1)))
#define AS3 __attribute__((address_space(3)))

#define IMG        384
#define NPTS       (IMG * IMG)        // 147456
#define CHUNK_C    512                // candidates staged in LDS per step (8KB)
#define NCHUNK     (NPTS / CHUNK_C)   // 288
#define LPT        (CHUNK_C / 256)    // b128 stage loads per thread per chunk = 2
#define WAVES_PB   8                  // 8 wave32 per block, one 16-query tile each
#define NQT        (NPTS / 16)        // 9216 query tiles per direction
#define NBLK       (NQT / WAVES_PB)   // 1152 blocks per direction

#if defined(__AMDGCN__) && __has_builtin(__builtin_amdgcn_global_load_async_to_lds_b128)
#define USE_ASYNC 1
#else
#define USE_ASYNC 0
#endif

#if defined(__AMDGCN__) && __has_builtin(__builtin_amdgcn_s_wait_asynccnt)
#define WAIT_ASYNC(n) __builtin_amdgcn_s_wait_asynccnt(n)
#elif defined(__AMDGCN__)
#define WAIT_ASYNC(n) asm volatile("s_wait_asynccnt " #n ::: "memory")
#else
#define WAIT_ASYNC(n)
#endif

// ---------------------------------------------------------------- utilities
__global__ void zero_acc_kernel(float* acc) {
    if (threadIdx.x < 2) acc[threadIdx.x] = 0.0f;
}

// ------------------------------------------------- depth -> extended clouds
// Pq/Qq: (x, y, z, 1)           -> WMMA A operand rows
// Pc/Qc: (-2x, -2y, -2z, |p|^2) -> WMMA B operand columns
__global__ void transform_kernel(const float* __restrict__ fake,
                                 const float* __restrict__ tar,
                                 const int* __restrict__ psh,
                                 const int* __restrict__ psw,
                                 float4* __restrict__ Pq, float4* __restrict__ Pc,
                                 float4* __restrict__ Qq, float4* __restrict__ Qc) {
    int i = blockIdx.x * blockDim.x + threadIdx.x;
    if (i >= NPTS) return;
    int r = i / IMG;
    int c = i - r * IMG;

    const float DEG2RAD = 0.017453292519943295f;
    float sh = (float)psh[0];
    float sw = (float)psw[0];
    float crop_w_deg = sw * (360.0f / 1285.0f);
    float crop_h_deg = sh * (123.5f / 438.0f);
    const float fh_crop = 360.0f * 384.0f / 1285.0f;
    const float fv_crop = 123.5f * 384.0f / 438.0f;

    float yaw   = (-fh_crop * (float)c * (1.0f / 384.0f) + crop_w_deg) * DEG2RAD;
    float pitch = (-fv_crop * (float)r * (1.0f / 384.0f) + crop_h_deg) * DEG2RAD;

    float sy = sinf(yaw),   cy = cosf(yaw);
    float sp = sinf(pitch), cp = cosf(pitch);

    float dt = tar[i];   // P = transform(tar)
    float df = fake[i];  // Q = transform(fake)

    float px = dt * sy * sp, py = dt * cy * sp, pz = dt * cp;
    float qx = df * sy * sp, qy = df * cy * sp, qz = df * cp;

    Pq[i] = make_float4(px, py, pz, 1.0f);
    Pc[i] = make_float4(-2.0f * px, -2.0f * py, -2.0f * pz, px*px + py*py + pz*pz);
    Qq[i] = make_float4(qx, qy, qz, 1.0f);
    Qc[i] = make_float4(-2.0f * qx, -2.0f * qy, -2.0f * qz, qx*qx + qy*qy + qz*qz);
}

// ------------------------------------------- WMMA brute-force nearest neighbor
// grid.y = direction: 0 -> queries P, candidates Q ; 1 -> queries Q, candidates P
// Each wave owns one 16-query tile; block double-buffers 512-candidate chunks
// in LDS via CDNA5 async global->LDS copies (ASYNCcnt).
__global__ void __launch_bounds__(256)
nn_min_kernel(const float4* __restrict__ Pq, const float4* __restrict__ Pc,
              const float4* __restrict__ Qq, const float4* __restrict__ Qc,
              float* __restrict__ acc) {
    __shared__ __align__(16) float sB[2][CHUNK_C * 4];

    const int dir  = blockIdx.y;
    const float4* Q4 = dir ? Qq : Pq;  // queries
    const float4* C4 = dir ? Pc : Qc;  // candidates

    const int tid  = threadIdx.x;
    const int lane = tid & 31;
    const int wave = tid >> 5;
    const int qtile = blockIdx.x * WAVES_PB + wave;

    // A operand: lane L holds query row (L%16), K-pair (L/16)*2  -> 64-bit load
    const int qrow = qtile * 16 + (lane & 15);
    const int k0   = (lane >> 4) * 2;
    const float* qp = (const float*)&Q4[qrow];
    v2f a;
    a.x = qp[k0];
    a.y = qp[k0 + 1];

    // partial |a|^2 per lane: lanes 0-15 hold (x,y), lanes 16-31 hold (z, 1)
    float partial = (lane < 16) ? (a.x * a.x + a.y * a.y) : (a.x * a.x);

    v8f mins;
#pragma unroll
    for (int v = 0; v < 8; ++v) mins[v] = 3.4e38f;

#if USE_ASYNC
    // prologue: stage chunk 0 into buffer 0
#pragma unroll
    for (int j = 0; j < LPT; ++j) {
        __builtin_amdgcn_global_load_async_to_lds_b128(
            (AS1 v4i*)(v4i*)&C4[j * 256 + tid],
            (AS3 v4i*)(v4i*)&sB[0][(j * 256 + tid) * 4],
            0, 0);
    }
    for (int ch = 0; ch < NCHUNK; ++ch) {
        const int cur = ch & 1;
        if (ch + 1 < NCHUNK) {
            // overwrite of sB[cur^1] is safe: end-of-iteration barrier of ch-1
#pragma unroll
            for (int j = 0; j < LPT; ++j) {
                __builtin_amdgcn_global_load_async_to_lds_b128(
                    (AS1 v4i*)(v4i*)&C4[(ch + 1) * CHUNK_C + j * 256 + tid],
                    (AS3 v4i*)(v4i*)&sB[cur ^ 1][(j * 256 + tid) * 4],
                    0, 0);
            }
            WAIT_ASYNC(LPT);   // chunk ch copies (older, in-order) complete
        } else {
            WAIT_ASYNC(0);
        }
        __syncthreads();       // whole buffer valid across all 8 waves

        const float* buf = sB[cur];
#pragma unroll 4
        for (int t = 0; t < CHUNK_C / 16; ++t) {
            const float* bp = buf + (t * 16 + (lane & 15)) * 4 + k0;
            v2f b = *(const v2f*)bp;                // ds_load_b64
            v8f cacc = {};
            // D[m][n] = |b_n|^2 - 2 a_m . b_n   (proxy distance, min-able)
            v8f d = __builtin_amdgcn_wmma_f32_16x16x4_f32(
                        false, a, false, b, (short)0, cacc, false, false);
#pragma unroll
            for (int v = 0; v < 8; ++v) mins[v] = fminf(mins[v], d[v]);
        }
        __syncthreads();       // readers done -> buffer reusable
    }
#else
    for (int ch = 0; ch < NCHUNK; ++ch) {
        __syncthreads();
#pragma unroll
        for (int j = 0; j < LPT; ++j)
            ((float4*)sB[0])[j * 256 + tid] = C4[ch * CHUNK_C + j * 256 + tid];
        __syncthreads();

        const float* buf = sB[0];
#pragma unroll 4
        for (int t = 0; t < CHUNK_C / 16; ++t) {
            const float* bp = buf + (t * 16 + (lane & 15)) * 4 + k0;
            v2f b = *(const v2f*)bp;
            v8f cacc = {};
            v8f d = __builtin_amdgcn_wmma_f32_16x16x4_f32(
                        false, a, false, b, (short)0, cacc, false, false);
#pragma unroll
            for (int v = 0; v < 8; ++v) mins[v] = fminf(mins[v], d[v]);
        }
    }
#endif

    // reduce over candidate columns: min across each 16-lane half
#pragma unroll
    for (int v = 0; v < 8; ++v) {
        float x = mins[v];
        x = fminf(x, __shfl_xor(x, 1, 16));
        x = fminf(x, __shfl_xor(x, 2, 16));
        x = fminf(x, __shfl_xor(x, 4, 16));
        x = fminf(x, __shfl_xor(x, 8, 16));
        mins[v] = x;
    }

    // lanes 0-7 emit rows 0-7 (VGPR v = lane), lanes 16-23 emit rows 8-15
    const int l7 = lane & 7;
    float sel = mins[0];
#pragma unroll
    for (int v = 1; v < 8; ++v) sel = (l7 == v) ? mins[v] : sel;

    const int row = (lane & 7) | ((lane & 16) >> 1);   // row within tile, 0..15
    float n2 = __shfl(partial, row, 32) + __shfl(partial, row + 16, 32); // |a_row|^2

    if ((lane & 8) == 0) {
        atomicAdd(&acc[dir], sel + n2);   // min squared distance for this row
    }
}

__global__ void finalize_kernel(const float* __restrict__ acc, float* __restrict__ out) {
    if (threadIdx.x == 0) out[0] = (acc[0] + acc[1]) * (1.0f / (float)NPTS);
}

// ----------------------------------------------------------------- launcher
extern "C" void kernel_launch(void* const* d_in, const int* in_sizes, int n_in,
                              void* d_out, int out_size, void* d_ws, size_t ws_size,
                              hipStream_t stream) {
    const float* fake = (const float*)d_in[0];
    const float* tar  = (const float*)d_in[1];
    const int*   sh   = (const int*)d_in[2];
    const int*   sw   = (const int*)d_in[3];
    float*       out  = (float*)d_out;

    char*   ws  = (char*)d_ws;
    float*  acc = (float*)ws;                 // 2 accumulators
    float4* Pq  = (float4*)(ws + 16);
    float4* Pc  = Pq + NPTS;
    float4* Qq  = Pc + NPTS;
    float4* Qc  = Qq + NPTS;                  // total ~9.4 MB of d_ws

    zero_acc_kernel<<<1, 32, 0, stream>>>(acc);
    transform_kernel<<<NPTS / 256, 256, 0, stream>>>(fake, tar, sh, sw, Pq, Pc, Qq, Qc);
    nn_min_kernel<<<dim3(NBLK, 2), 256, 0, stream>>>(Pq, Pc, Qq, Qc, acc);
    finalize_kernel<<<1, 32, 0, stream>>>(acc, out);
}